// dca_layer_94489280941
// MI455X (gfx1250) — compile-verified
//
#include <hip/hip_runtime.h>
#include <stdint.h>

// Problem constants (from the reference: B=32, C=512, H=W=56, K=3)
#define B_   32
#define C_   512
#define HW_  3136           // 56*56 contiguous floats per (b,c) plane
#define HW4_ 784            // HW_/4 float4 chunks per plane
#define BLK  256

typedef float v4f __attribute__((ext_vector_type(4)));

// Generic pointer to a __shared__ object: low 32 bits are the wave-relative
// LDS byte address; that is exactly what the
// async-to-LDS VDST operand wants.
__device__ __forceinline__ uint32_t lds_byte_addr(const void* p) {
    return (uint32_t)(uintptr_t)p;
}

// ---------------------------------------------------------------------------
// Kernel 1: global average pool over each (b,c) 56x56 plane.
// Stages the plane into LDS via the gfx1250 async global->LDS path
// (ASYNCcnt), then reduces with b128 LDS reads + tree reduction.
// ---------------------------------------------------------------------------
__global__ __launch_bounds__(BLK) void pool_kernel(const float* __restrict__ x,
                                                   float* __restrict__ pooled) {
    __shared__ __align__(16) float tile[HW_];   // 12544 B
    __shared__ float red[BLK];

    const int bc = blockIdx.x;
    const int t  = threadIdx.x;
    const float* src = x + (size_t)bc * HW_;

    // Async copy: 16 bytes per lane per issue. Loop tail leaves some lanes
    // inactive (per-lane EXEC gating on the async op).
    for (int i = t; i < HW4_; i += BLK) {
        uint32_t laddr  = lds_byte_addr(&tile[i * 4]);
        const float* g  = src + (size_t)i * 4;
        asm volatile("global_load_async_to_lds_b128 %0, %1, off"
                     :: "v"(laddr), "v"(g)
                     : "memory");
    }
    // Wait for this wave's async transfers, then cross-wave barrier.
    asm volatile("s_wait_asynccnt 0" ::: "memory");
    __syncthreads();

    float s = 0.0f;
    const v4f* t4 = reinterpret_cast<const v4f*>(tile);
    for (int i = t; i < HW4_; i += BLK) {
        v4f v = t4[i];                          // ds_load_b128
        s += (v.x + v.y) + (v.z + v.w);
    }
    red[t] = s;
    __syncthreads();
    #pragma unroll
    for (int stride = BLK / 2; stride > 0; stride >>= 1) {
        if (t < stride) red[t] += red[t + stride];
        __syncthreads();
    }
    if (t == 0) pooled[bc] = red[0] * (1.0f / (float)HW_);
}

// ---------------------------------------------------------------------------
// Kernel 2: deformable 1-D conv over channels of the pooled vector -> gate.
// One block per batch, one thread per channel; pooled row lives in LDS for
// the data-dependent bilinear gathers.
// ---------------------------------------------------------------------------
__global__ __launch_bounds__(C_) void gate_kernel(const float* __restrict__ pooled,
                                                  const float* __restrict__ w_offset,  // (3,1,3)
                                                  const float* __restrict__ w_deform,  // (1,1,3)
                                                  const float* __restrict__ b_deform,  // (1,)
                                                  float* __restrict__ gate) {
    __shared__ float row[C_];
    const int b = blockIdx.x;
    const int c = threadIdx.x;

    row[c] = pooled[b * C_ + c];
    __syncthreads();

    // Zero-padded conv input neighbors (lax.conv_general_dilated pads with 0).
    const float pm1 = (c >= 1)      ? row[c - 1] : 0.0f;
    const float p00 = row[c];
    const float pp1 = (c + 1 < C_)  ? row[c + 1] : 0.0f;

    float out = b_deform[0];
    #pragma unroll
    for (int k = 0; k < 3; ++k) {
        // Cross-correlation: offset[b,k,c] = sum_j pooled[b, c+j-1] * w_offset[k,0,j]
        float off = fmaf(w_offset[k * 3 + 0], pm1,
                    fmaf(w_offset[k * 3 + 1], p00,
                         w_offset[k * 3 + 2] * pp1));
        float p = (float)(c + k - 1) + off;
        p = fminf(fmaxf(p, 0.0f), (float)(C_ - 1));
        float pf = floorf(p);
        float fr = p - pf;
        int i0 = (int)pf;
        int i1 = min(i0 + 1, C_ - 1);
        float samp = fmaf(row[i0], 1.0f - fr, row[i1] * fr);
        out = fmaf(samp, w_deform[k], out);
    }
    gate[b * C_ + c] = 1.0f / (1.0f + __expf(-out));
}

// ---------------------------------------------------------------------------
// Kernel 3: out = x * gate (gate broadcast over the 56x56 plane).
// gate index is block-uniform -> scalar load; x streamed as b128; output
// stored non-temporally (written once, never re-read).
// ---------------------------------------------------------------------------
__global__ __launch_bounds__(BLK) void scale_kernel(const float* __restrict__ x,
                                                    const float* __restrict__ gate,
                                                    float* __restrict__ out) {
    const int bc  = blockIdx.x;
    const float g = gate[bc];                              // uniform -> s_load
    const v4f* xs = reinterpret_cast<const v4f*>(x) + (size_t)bc * HW4_;
    v4f*       os = reinterpret_cast<v4f*>(out)    + (size_t)bc * HW4_;

    for (int i = threadIdx.x; i < HW4_; i += BLK) {
        v4f v = xs[i];
        v4f r = { v.x * g, v.y * g, v.z * g, v.w * g };
        __builtin_nontemporal_store(r, &os[i]);            // global_store_b128 NT
    }
}

// ---------------------------------------------------------------------------
// Launch: d_in order = {x, w_offset, w_deform, b_deform}; d_out = (B,C,H,W) f32.
// Workspace layout: pooled[B*C] | gate[B*C]  (128 KB total).
// ---------------------------------------------------------------------------
extern "C" void kernel_launch(void* const* d_in, const int* in_sizes, int n_in,
                              void* d_out, int out_size, void* d_ws, size_t ws_size,
                              hipStream_t stream) {
    const float* x        = (const float*)d_in[0];
    const float* w_offset = (const float*)d_in[1];
    const float* w_deform = (const float*)d_in[2];
    const float* b_deform = (const float*)d_in[3];
    float* out    = (float*)d_out;
    float* pooled = (float*)d_ws;            // B_*C_ floats
    float* gate   = pooled + B_ * C_;        // B_*C_ floats

    pool_kernel <<<B_ * C_, BLK, 0, stream>>>(x, pooled);
    gate_kernel <<<B_,      C_,  0, stream>>>(pooled, w_offset, w_deform, b_deform, gate);
    scale_kernel<<<B_ * C_, BLK, 0, stream>>>(x, gate, out);
}